// forward_warp_24859270709537
// MI455X (gfx1250) — compile-verified
//
#include <hip/hip_runtime.h>
#include <stdint.h>

// Problem constants from the reference: B=4, C=16, H=512, W=512
#define BB 4
#define CC 16
#define HH 512
#define WW 512
#define HWP (HH * WW)            // 262144 elements per plane
#define NPIX (BB * HH * WW)      // 1,048,576 source pixels

typedef float v2f __attribute__((ext_vector_type(2)));
typedef float v4f __attribute__((ext_vector_type(4)));

// Non-returning f32 global atomic add with a compile-time instruction offset
// (signed 24-bit, +-8MiB). One 64-bit base address serves 8 channel planes
// x 4 corners purely via immediates -> minimal VALU address math and VGPR
// pressure on the atomic-issue path. Resolves in L2's atomic units, tracked
// with STOREcnt (implicitly waited at s_endpgm).
__device__ __forceinline__ void atomic_add_f32_off(float* base, int imm_off, float v) {
    asm volatile("global_atomic_add_f32 %0, %1, off offset:%2"
                 :: "v"((uint64_t)(uintptr_t)base), "v"(v), "i"(imm_off)
                 : "memory");
}

__global__ __launch_bounds__(256) void zero_out_kernel(v4f* __restrict__ out, int n4) {
    int i = blockIdx.x * 256 + threadIdx.x;
    if (i < n4) out[i] = (v4f){0.f, 0.f, 0.f, 0.f};
}

__global__ __launch_bounds__(256) void fwarp_kernel(const float* __restrict__ im0,
                                                    const v2f* __restrict__ flow,
                                                    float* __restrict__ out) {
    int p = blockIdx.x * 256 + threadIdx.x;      // pixel id in [0, B*H*W)
    int b  = p >> 18;                            // / (H*W)
    int hw = p & (HWP - 1);
    int h  = hw >> 9;                            // / W
    int w  = hw & (WW - 1);

    // flow[b,h,w,{x,y}] : one coalesced b64 NT load per pixel (streamed once)
    v2f f = __builtin_nontemporal_load(&flow[p]);

    float xd = (float)w + f.x;
    float yd = (float)h + f.y;
    float xff = floorf(xd);
    float yff = floorf(yd);
    int ix = (int)xff;
    int iy = (int)yff;

    // reference: contribution only when ALL four corners are in-bounds
    if (ix < 0 || iy < 0 || ix >= WW - 1 || iy >= HH - 1) return;

    float tx = xd - xff;
    float ty = yd - yff;
    float w_nw = (1.f - tx) * (1.f - ty);
    float w_ne = tx * (1.f - ty);
    float w_sw = (1.f - tx) * ty;
    float w_se = tx * ty;

    const float* src  = im0 + (size_t)b * (CC * HWP) + hw;            // plane-strided reads
    float*       dst0 = out + (size_t)b * (CC * HWP) + iy * WW + ix;  // NW corner, channels 0-7
    float*       dst1 = dst0 + 8 * HWP;                               // NW corner, channels 8-15

    // pull the far half of the channel stripe while early atomics issue
    __builtin_prefetch(src + 8 * HWP, 0, 0);

#pragma unroll
    for (int c = 0; c < 8; ++c) {
        // read-once data: NT hint keeps L2 capacity for the atomic-hot output
        float v0 = __builtin_nontemporal_load(src + c * HWP);
        float v1 = __builtin_nontemporal_load(src + (c + 8) * HWP);
        const int o = c * HWP * 4;               // byte offset of channel plane (< 8 MiB)

        atomic_add_f32_off(dst0, o,                v0 * w_nw);
        atomic_add_f32_off(dst0, o + 4,            v0 * w_ne);
        atomic_add_f32_off(dst0, o + 4 * WW,       v0 * w_sw);
        atomic_add_f32_off(dst0, o + 4 * WW + 4,   v0 * w_se);

        atomic_add_f32_off(dst1, o,                v1 * w_nw);
        atomic_add_f32_off(dst1, o + 4,            v1 * w_ne);
        atomic_add_f32_off(dst1, o + 4 * WW,       v1 * w_sw);
        atomic_add_f32_off(dst1, o + 4 * WW + 4,   v1 * w_se);
    }
}

extern "C" void kernel_launch(void* const* d_in, const int* in_sizes, int n_in,
                              void* d_out, int out_size, void* d_ws, size_t ws_size,
                              hipStream_t stream) {
    const float* im0  = (const float*)d_in[0];   // [B,C,H,W] f32
    const v2f*   flow = (const v2f*)d_in[1];     // [B,H,W,2] f32
    float*       out  = (float*)d_out;           // [B,C,H,W] f32

    // 1) zero the (poisoned) accumulator
    int n4 = out_size / 4;                       // 4,194,304 float4 stores
    zero_out_kernel<<<(n4 + 255) / 256, 256, 0, stream>>>((v4f*)out, n4);

    // 2) bilinear forward splat, 4 non-returning L2 atomics per channel-pixel
    fwarp_kernel<<<NPIX / 256, 256, 0, stream>>>(im0, flow, out);
}